// Condensation_47897475285619
// MI455X (gfx1250) — compile-verified
//
#include <hip/hip_runtime.h>

// MI455X (gfx1250) fused rain-drop blur.
// Roofline: 10 serial drops over a 113MB fp32 image. Fused per-drop kernel
// => ~226MB HBM traffic/drop => ~100us floor @ 23.3TB/s (halo redundancy is
// absorbed by the 192MB L2, which holds the whole image). A VALU-only blur
// (~26 GFLOP) would exceed that floor (~170us), so both separable conv
// passes run as banded-matrix matmuls on V_WMMA_F32_16X16X4_F32, making the
// kernel memory-bound. Kernels are templated on ksize so the K-chunk loops
// fully unroll and the banded-weight fragments live in fixed VGPRs (no
// m0/v_movrels indexing on the WMMA critical path).

typedef float v2f __attribute__((ext_vector_type(2)));
typedef float v8f __attribute__((ext_vector_type(8)));

#define IMG_H 768
#define IMG_W 768
#define TILE  64

__device__ __forceinline__ int reflect_idx(int i, int n) {
  if (i < 0) i = -i;
  else if (i >= n) i = 2 * n - 2 - i;
  return i;
}

// Pass 1 per drop: materialize the batch/channel-invariant HxW mask once
// (2.36MB, stays hot in L2) instead of 48x redundant pow/exp per pixel.
__global__ __launch_bounds__(256)
void mask_kernel(float* __restrict__ maskbuf,
                 const float* __restrict__ pos,
                 const float* __restrict__ rad, int j) {
  int t = blockIdx.x * 256 + threadIdx.x;
  if (t >= IMG_H * IMG_W) return;
  int h = t / IMG_W, w = t - (t / IMG_W) * IMG_W;
  float px = fminf(fmaxf(pos[2 * j + 0], -1.f), 1.f);
  float py = fminf(fmaxf(pos[2 * j + 1], -1.f), 1.f);
  float wr = fminf(fmaxf(rad[j], 60.f), 80.f);
  float hr = wr * 0.8f;
  float x0 = (px + 1.f) * 0.5f * (float)IMG_W;
  float y0 = (py + 1.f) * 0.5f * (float)IMG_H;
  float dy = ((float)h - y0) / hr;
  float dx = ((float)w - x0) / wr;
  float d  = dy * dy + dx * dx;
  float m  = expf(-powf(d, 1.8f) + 1e-10f);
  maskbuf[t] = fminf(fmaxf(m, 0.f), 1.f);
}

// Fused per-drop kernel: one block = one 64x64 tile of one (b,c) plane.
// LDS holds raw tile + mask tile (reflect halo + zero margin) and the
// vertical-blur intermediate. Both 1-D conv passes are 16x16-tile banded
// matmuls: D(16x16) = Wband(16xKpad) x patch / patch x Wband.
template <int K>
__global__ __launch_bounds__(256)
void blur_kernel(const float* __restrict__ src, float* __restrict__ dst,
                 const float* __restrict__ maskbuf, float sigma) {
  constexpr int P   = K / 2;              // pad
  constexpr int E   = TILE + 2 * P;       // valid halo extent
  constexpr int R   = E + 4;              // loaded extent incl. zero margin
  constexpr int KC  = (K + 18) / 4;       // ceil((K+15)/4) K-chunks (<=13)
  constexpr int NGX = (48 + 4 * KC + 15) / 16;  // x-groups for v-pass
  constexpr int VJOBS = 4 * NGX;
  // s_in/s_mask stride: must cover both the loaded extent (cols < R) and
  // the WMMA x-group reads (cols < 16*NGX).
  constexpr int SST = (R > 16 * NGX) ? R : 16 * NGX;
  constexpr int SVS = 16 * NGX + 1;       // odd stride: conflict-free cols
  static_assert(SST >= R && SST >= 16 * NGX, "stride too small");
  static_assert(48 + 4 * KC <= 16 * NGX, "h-pass reads beyond v-pass cover");
  static_assert(48 + 4 * KC - 1 < R, "h-pass would touch poisoned columns");

  __shared__ float s_mask[R * SST];
  __shared__ float s_in  [R * SST];
  __shared__ float s_vert[TILE * SVS];
  __shared__ float s_w[K];

  const int tid   = threadIdx.x;
  const int lane  = tid & 31;
  const int wid   = tid >> 5;
  const int l16   = lane & 15;
  const int lhalf = lane >> 4;

  const int h0 = blockIdx.y * TILE;
  const int w0 = blockIdx.x * TILE;
  const size_t plane = (size_t)blockIdx.z * (size_t)(IMG_H * IMG_W);

  // --- Gaussian weights ---
  if (tid < K) {
    float half = (float)(K - 1) * 0.5f;
    float t = ((float)tid - half) / sigma;
    s_w[tid] = expf(-0.5f * t * t);
  }
  __syncthreads();
  if (tid == 0) {
    float s = 0.f;
    for (int t = 0; t < K; ++t) s += s_w[t];
    float inv = 1.f / s;
    for (int t = 0; t < K; ++t) s_w[t] *= inv;
  }

  // --- mask tile with reflect halo + zero margin (loaded once) ---
  for (int idx = tid; idx < R * R; idx += 256) {
    int y = idx / R, x = idx - y * R;
    float v = 0.f;
    if (y < E && x < E) {
      int gh = reflect_idx(h0 - P + y, IMG_H);
      int gw = reflect_idx(w0 - P + x, IMG_W);
      v = maskbuf[gh * IMG_W + gw];
    }
    s_mask[y * SST + x] = v;
  }
  __syncthreads();  // also publishes normalized weights

  // --- per-lane constant banded-weight fragments (fixed VGPRs) ---
  // Vertical A operand W[m][i] = w[i-m] and horizontal B operand
  // Wh[i][n] = w[i-n] yield identical per-lane values under the 16x4 f32
  // layout (K split {0,1}/{2,3} across lane halves).
  v2f wA[KC];
#pragma unroll
  for (int kk = 0; kk < KC; ++kk) {
    int i0 = 4 * kk + 2 * lhalf;
    int d0 = i0 - l16;
    int d1 = i0 + 1 - l16;
    wA[kk].x = (d0 >= 0 && d0 < K) ? s_w[d0] : 0.f;
    wA[kk].y = (d1 >= 0 && d1 < K) ? s_w[d1] : 0.f;
  }

  // --- load raw source tile (reflect halo, zero margin) ---
  const float* sp = src + plane;
  for (int idx = tid; idx < R * R; idx += 256) {
    int y = idx / R, x = idx - y * R;
    float v = 0.f;
    if (y < E && x < E) {
      int gh = reflect_idx(h0 - P + y, IMG_H);
      int gw = reflect_idx(w0 - P + x, IMG_W);
      v = sp[gh * IMG_W + gw];
    }
    s_in[y * SST + x] = v;
  }
  __syncthreads();

  // --- vertical blur of (src*mask): D = Wband(16xKpad) x patch(Kpadx16) ---
  for (int job = wid; job < VJOBS; job += 8) {
    int m0 = (job & 3) * 16;
    int x0 = (job >> 2) * 16;
    v8f acc = {0.f, 0.f, 0.f, 0.f, 0.f, 0.f, 0.f, 0.f};
#pragma unroll
    for (int kk = 0; kk < KC; ++kk) {
      int r0 = m0 + 4 * kk + 2 * lhalf;
      int c  = x0 + l16;
      v2f bfrag;
      bfrag.x = s_in[ r0      * SST + c] * s_mask[ r0      * SST + c];
      bfrag.y = s_in[(r0 + 1) * SST + c] * s_mask[(r0 + 1) * SST + c];
      acc = __builtin_amdgcn_wmma_f32_16x16x4_f32(
          false, wA[kk], false, bfrag, (short)0, acc, false, false);
    }
#pragma unroll
    for (int v = 0; v < 8; ++v) {
      int m = m0 + v + 8 * lhalf;
      s_vert[m * SVS + x0 + l16] = acc[v];
    }
  }
  __syncthreads();

  // --- horizontal blur: D = patch(16xKpad) x Wband(Kpadx16); blend; store ---
  float* dp = dst + plane;
  for (int job = wid; job < 16; job += 8) {
    int m0 = (job >> 2) * 16;
    int n0 = (job & 3) * 16;
    v8f acc = {0.f, 0.f, 0.f, 0.f, 0.f, 0.f, 0.f, 0.f};
#pragma unroll
    for (int kk = 0; kk < KC; ++kk) {
      int r = m0 + l16;
      int c = n0 + 4 * kk + 2 * lhalf;
      v2f afrag;
      afrag.x = s_vert[r * SVS + c];
      afrag.y = s_vert[r * SVS + c + 1];
      acc = __builtin_amdgcn_wmma_f32_16x16x4_f32(
          false, afrag, false, wA[kk], (short)0, acc, false, false);
    }
#pragma unroll
    for (int v = 0; v < 8; ++v) {
      int m = m0 + v + 8 * lhalf;
      int n = n0 + l16;
      float mk  = s_mask[(P + m) * SST + (P + n)];
      float raw = s_in  [(P + m) * SST + (P + n)];
      dp[(h0 + m) * IMG_W + (w0 + n)] = acc[v] * mk + raw * (1.f - mk);
    }
  }
}

template <int K>
static inline void launch_blur(const float* src, float* dst,
                               const float* maskbuf, float sigma,
                               hipStream_t stream) {
  blur_kernel<K><<<dim3(IMG_W / TILE, IMG_H / TILE, 48), 256, 0, stream>>>(
      src, dst, maskbuf, sigma);
}

extern "C" void kernel_launch(void* const* d_in, const int* in_sizes, int n_in,
                              void* d_out, int out_size, void* d_ws, size_t ws_size,
                              hipStream_t stream) {
  const float* img = (const float*)d_in[0];   // [16,3,768,768]
  const float* pos = (const float*)d_in[1];   // [10,2]
  const float* rad = (const float*)d_in[2];   // [10]
  float* out = (float*)d_out;

  const size_t imgElems = (size_t)16 * 3 * IMG_H * IMG_W;
  float* ws0     = (float*)d_ws;              // ping buffer (113 MB)
  float* maskbuf = ws0 + imgElems;            // per-drop HxW mask (2.4 MB)

  static const int   KS[10]  = {23, 37, 13, 23, 13, 33, 25, 27, 15, 21};
  static const float SIG[10] = {11.3535f, 17.9381f, 5.7966f, 10.8586f, 5.5301f,
                                15.9075f, 12.3225f, 13.4871f, 6.6639f, 9.5413f};

  const int maskBlocks = (IMG_H * IMG_W + 255) / 256;

  const float* src = img;
  for (int j = 0; j < 10; ++j) {
    float* dst = (j & 1) ? out : ws0;  // 10 hops: drop 9 (odd) lands in d_out
    mask_kernel<<<maskBlocks, 256, 0, stream>>>(maskbuf, pos, rad, j);
    float sg = SIG[j];
    switch (KS[j]) {
      case 13: launch_blur<13>(src, dst, maskbuf, sg, stream); break;
      case 15: launch_blur<15>(src, dst, maskbuf, sg, stream); break;
      case 21: launch_blur<21>(src, dst, maskbuf, sg, stream); break;
      case 23: launch_blur<23>(src, dst, maskbuf, sg, stream); break;
      case 25: launch_blur<25>(src, dst, maskbuf, sg, stream); break;
      case 27: launch_blur<27>(src, dst, maskbuf, sg, stream); break;
      case 33: launch_blur<33>(src, dst, maskbuf, sg, stream); break;
      case 37: launch_blur<37>(src, dst, maskbuf, sg, stream); break;
      default: break;  // unreachable: KS only contains the cases above
    }
    src = dst;
  }
}